// LSTMClassifier_3788161155726
// MI455X (gfx1250) — compile-verified
//
#include <hip/hip_runtime.h>
#include <math.h>

// ---------------------------------------------------------------------------
// Types for WMMA fragments (fp32 16x16x4 path: exact f32, matches reference)
// ---------------------------------------------------------------------------
typedef float v2f __attribute__((ext_vector_type(2)));
typedef float v8f __attribute__((ext_vector_type(8)));

#define N_NODES 16384
#define FEAT    512
#define EMB     256
#define HID     512
#define GATES   2048   // 4*HID
#define NWG_LSTM 32    // persistent workgroups for the sequential scan

// gfx1250 async global->LDS copies (ASYNCcnt-tracked, no VGPR round trip)
#if defined(__has_builtin)
#if __has_builtin(__builtin_amdgcn_global_load_async_to_lds_b128) && \
    __has_builtin(__builtin_amdgcn_s_wait_asynccnt)
#define HAVE_ASYNC_LDS 1
#endif
#endif

#ifdef HAVE_ASYNC_LDS
// payload type per clang diagnostic: int __attribute__((vector_size(16)))
typedef int av4i __attribute__((vector_size(16)));
typedef __attribute__((address_space(1))) av4i* gas_v4i;
typedef __attribute__((address_space(3))) av4i* las_v4i;
#endif

// ---------------------------------------------------------------------------
// zero helper (float4 granularity)
// ---------------------------------------------------------------------------
__global__ void zero_f4(float4* __restrict__ p, long n4) {
    long i = (long)blockIdx.x * blockDim.x + threadIdx.x;
    if (i < n4) p[i] = make_float4(0.f, 0.f, 0.f, 0.f);
}

// ---------------------------------------------------------------------------
// node_emb = W_emb[tag] + b_emb  ->  x[:, 0:256]
// ---------------------------------------------------------------------------
__global__ void gather_emb(const int* __restrict__ tags,
                           const float* __restrict__ W_emb,
                           const float* __restrict__ b_emb,
                           float* __restrict__ x) {
    int i = blockIdx.x * blockDim.x + threadIdx.x;   // over N*64 float4
    if (i >= N_NODES * (EMB / 4)) return;
    int n  = i >> 6;
    int c4 = i & 63;
    int tag = tags[n];
    float4 w = ((const float4*)(W_emb + (size_t)tag * EMB))[c4];
    float4 b = ((const float4*)b_emb)[c4];
    float4 o = make_float4(w.x + b.x, w.y + b.y, w.z + b.z, w.w + b.w);
    ((float4*)(x + (size_t)n * 512))[c4] = o;
}

// ---------------------------------------------------------------------------
// neighbor-tag histogram: hist[nid, tag] += 1  (hardware fp32 atomic)
// ---------------------------------------------------------------------------
__global__ void edge_hist(const int* __restrict__ nid,
                          const int* __restrict__ tag,
                          float* __restrict__ hist, int E) {
    int e = blockIdx.x * blockDim.x + threadIdx.x;
    if (e < E) {
        unsafeAtomicAdd(&hist[(size_t)nid[e] * FEAT + tag[e]], 1.0f);
    }
}

// ---------------------------------------------------------------------------
// fp32 WMMA GEMM:  Out[m,n] = sum_k A[m,k] * Bsrc(k,n) + bias[n]
//   A: [M,512] row-major (K fixed at 512)
//   BTRANS=true : Bsrc(k,n) = B[n*512 + k]   (x @ W_ih^T)
//   BTRANS=false: Bsrc(k,n) = B[k*N   + n]   (hist @ W_emb)
// WG = 256 thr (8 waves). Tile: 32 M-rows x 128 N-cols.
// Each wave: 32x16 output = TWO 16x16 accumulators sharing one B fragment
// (halves B traffic, back-to-back WMMA issue). A tile staged in padded
// dynamic LDS (66KB) via async global->LDS when available.
// ---------------------------------------------------------------------------
#define AS_LD 516   // 512 + 4 pad: conflict-free 16-lane row reads, 16B rows

template <bool BTRANS>
__global__ void wmma_gemm_f32(const float* __restrict__ A,
                              const float* __restrict__ B,
                              const float* __restrict__ bias,
                              float* __restrict__ Out,
                              int ldo, int N) {
    const int K = 512;
    extern __shared__ float as[];               // 32 x AS_LD floats
    int tid  = threadIdx.x;
    int wave = tid >> 5;
    int lane = tid & 31;
    int Mbase = blockIdx.x * 32;

    // stage A tile: 32 rows x 512 floats (16B-aligned rows: AS_LD*4 = 2064)
#ifdef HAVE_ASYNC_LDS
    for (int i = tid; i < 32 * 128; i += 256) {
        int r  = i >> 7;
        int c4 = i & 127;
        __builtin_amdgcn_global_load_async_to_lds_b128(
            (gas_v4i)(A + (size_t)(Mbase + r) * K + c4 * 4),
            (las_v4i)(&as[r * AS_LD + c4 * 4]),
            0, 0);
    }
    __builtin_amdgcn_s_wait_asynccnt(0);
#else
    for (int i = tid; i < 32 * 128; i += 256) {
        int r  = i >> 7;
        int c4 = i & 127;
        float4 v = ((const float4*)(A + (size_t)(Mbase + r) * K))[c4];
        ((float4*)(&as[r * AS_LD]))[c4] = v;
    }
#endif
    __syncthreads();

    int nBase = blockIdx.y * 128 + wave * 16;
    int mIdx  = lane & 15;           // A row within half-tile
    int kHalf = (lane >> 4) * 2;     // lanes 0-15 -> K {0,1}; 16-31 -> K {2,3}
    int nIdx  = nBase + (lane & 15); // B column

    const float* a0p = as + mIdx * AS_LD + kHalf;         // rows 0..15
    const float* a1p = as + (mIdx + 16) * AS_LD + kHalf;  // rows 16..31

    v8f acc0 = {};
    v8f acc1 = {};
    for (int k0 = 0; k0 < K; k0 += 4) {
        v2f a0, a1, b;
        a0.x = a0p[k0];
        a0.y = a0p[k0 + 1];
        a1.x = a1p[k0];
        a1.y = a1p[k0 + 1];
        if (BTRANS) {
            const float* bp = B + (size_t)nIdx * K + k0 + kHalf;
            if ((k0 & 63) == 0) __builtin_prefetch(bp + 64, 0, 0);
            b.x = bp[0];
            b.y = bp[1];
        } else {
            b.x = B[(size_t)(k0 + kHalf) * N + nIdx];
            b.y = B[(size_t)(k0 + kHalf + 1) * N + nIdx];
        }
        acc0 = __builtin_amdgcn_wmma_f32_16x16x4_f32(
            false, a0, false, b, (short)0, acc0, false, false);
        acc1 = __builtin_amdgcn_wmma_f32_16x16x4_f32(
            false, a1, false, b, (short)0, acc1, false, false);
    }

    // C/D layout: VGPR v -> row (v + 8*(lane>>4)), col (lane&15)
    int mOff = (lane >> 4) * 8;
    float bv = bias[nIdx];
#pragma unroll
    for (int v = 0; v < 8; ++v) {
        Out[(size_t)(Mbase + v + mOff) * ldo + nIdx]      = acc0[v] + bv;
        Out[(size_t)(Mbase + 16 + v + mOff) * ldo + nIdx] = acc1[v] + bv;
    }
}

// ---------------------------------------------------------------------------
// Sequential LSTM scan: persistent kernel, NWG_LSTM WGs x 256 threads.
// WG w owns h/c elements j in [w*16, w*16+16) and the 64 gate rows
// {g*512 + j}, whose W_hh slice (64x512 f32 = 128KB) lives in dynamic LDS.
// One monotonic release/acquire atomic barrier per step; h double-buffered
// in L2. hsum accumulated locally -> embed.
// ---------------------------------------------------------------------------
__global__ void lstm_scan(const float* __restrict__ pre,
                          const float* __restrict__ Whh,
                          float* __restrict__ hbuf,        // 2 x 512
                          float* __restrict__ embed,       // 512
                          unsigned* __restrict__ bar,
                          int nSteps) {
    extern __shared__ float sm[];
    float* Wl      = sm;                  // 64 * 513 (pad 1 -> no conflicts)
    float* hsh     = Wl + 64 * 513;       // 512
    float* gatesS  = hsh + 512;           // 64
    float* partial = gatesS + 64;         // 4 * 64
    float* cst     = partial + 256;       // 16
    float* hsumS   = cst + 16;            // 16

    int w   = blockIdx.x;
    int tid = threadIdx.x;

    // load the 64 gate rows for this WG into LDS
    for (int i = tid; i < 64 * 512; i += 256) {
        int r = i >> 9;
        int k = i & 511;
        int grow = (r >> 4) * HID + w * 16 + (r & 15);
        Wl[r * 513 + k] = Whh[(size_t)grow * HID + k];
    }
    if (tid < 16) { cst[tid] = 0.f; hsumS[tid] = 0.f; }
    __syncthreads();

    int part = tid >> 6;       // 0..3 (K quarter)
    int row  = tid & 63;       // 0..63 (gate row)

    for (int t = 0; t < nSteps; ++t) {
        // broadcast-load h_{t-1} (2KB) from the active buffer
        {
            float2 v = ((const float2*)(hbuf + (t & 1) * HID))[tid];
            hsh[2 * tid]     = v.x;
            hsh[2 * tid + 1] = v.y;
        }
        __syncthreads();

        // 64 dot products of length 512, 4 threads per row
        {
            const float* wr = Wl + row * 513 + part * 128;
            const float* hr = hsh + part * 128;
            float s = 0.f;
#pragma unroll 8
            for (int k = 0; k < 128; ++k) s = fmaf(wr[k], hr[k], s);
            partial[part * 64 + row] = s;
        }
        __syncthreads();

        if (tid < 64) {
            int gate = tid >> 4, jj = tid & 15;
            int grow = gate * HID + w * 16 + jj;
            gatesS[tid] = partial[tid] + partial[64 + tid] +
                          partial[128 + tid] + partial[192 + tid] +
                          pre[(size_t)t * GATES + grow];
        }
        __syncthreads();

        if (tid < 16) {
            float ig = 1.f / (1.f + expf(-gatesS[tid]));
            float fg = 1.f / (1.f + expf(-gatesS[16 + tid]));
            float gg = tanhf(gatesS[32 + tid]);
            float og = 1.f / (1.f + expf(-gatesS[48 + tid]));
            float c  = fg * cst[tid] + ig * gg;
            cst[tid] = c;
            float h  = og * tanhf(c);
            hsumS[tid] += h;
            hbuf[((t + 1) & 1) * HID + w * 16 + tid] = h;
            __threadfence();   // make h visible at device scope before arrive
        }
        __syncthreads();

        if (tid == 0) {
            __hip_atomic_fetch_add(bar, 1u, __ATOMIC_RELEASE,
                                   __HIP_MEMORY_SCOPE_AGENT);
            unsigned want = (unsigned)(t + 1) * NWG_LSTM;
            while (__hip_atomic_load(bar, __ATOMIC_ACQUIRE,
                                     __HIP_MEMORY_SCOPE_AGENT) < want) {
                __builtin_amdgcn_s_sleep(1);
            }
        }
        __syncthreads();
    }

    if (tid < 16) embed[w * 16 + tid] = hsumS[tid];
}

// ---------------------------------------------------------------------------
// MLP head: relu(embed @ W1 + b1) @ W2 + b2 -> log_softmax (10 outputs)
// ---------------------------------------------------------------------------
__global__ void mlp_head(const float* __restrict__ embed,
                         const float* __restrict__ W1,
                         const float* __restrict__ b1,
                         const float* __restrict__ W2,
                         const float* __restrict__ b2,
                         float* __restrict__ out) {
    __shared__ float e[512], h1[512], lg[10], red[2];
    int tid = threadIdx.x;   // 256 threads
    e[tid]       = embed[tid];
    e[tid + 256] = embed[tid + 256];
    __syncthreads();
#pragma unroll
    for (int rep = 0; rep < 2; ++rep) {
        int j = tid + rep * 256;
        float s = b1[j];
        for (int k = 0; k < 512; ++k) s = fmaf(e[k], W1[k * 512 + j], s);
        h1[j] = fmaxf(s, 0.f);
    }
    __syncthreads();
    if (tid < 10) {
        float s = b2[tid];
        for (int j = 0; j < 512; ++j) s = fmaf(h1[j], W2[j * 10 + tid], s);
        lg[tid] = s;
    }
    __syncthreads();
    if (tid == 0) {
        float m = lg[0];
        for (int c = 1; c < 10; ++c) m = fmaxf(m, lg[c]);
        float ss = 0.f;
        for (int c = 0; c < 10; ++c) ss += expf(lg[c] - m);
        red[0] = m;
        red[1] = logf(ss);
    }
    __syncthreads();
    if (tid < 10) out[tid] = lg[tid] - red[0] - red[1];
}

// ---------------------------------------------------------------------------
// host launcher
// ---------------------------------------------------------------------------
extern "C" void kernel_launch(void* const* d_in, const int* in_sizes, int n_in,
                              void* d_out, int out_size, void* d_ws, size_t ws_size,
                              hipStream_t stream) {
    (void)n_in; (void)out_size; (void)ws_size;
    const int*   node_tags = (const int*)d_in[0];
    const int*   nbr_nid   = (const int*)d_in[1];
    const int*   nbr_tag   = (const int*)d_in[2];
    const float* W_emb     = (const float*)d_in[4];
    const float* b_emb     = (const float*)d_in[5];
    const float* W_ih      = (const float*)d_in[6];
    const float* W_hh      = (const float*)d_in[7];
    const float* b_lstm    = (const float*)d_in[8];
    const float* W1        = (const float*)d_in[9];
    const float* b1        = (const float*)d_in[10];
    const float* W2        = (const float*)d_in[11];
    const float* b2        = (const float*)d_in[12];
    const int E = in_sizes[1];

    // workspace layout (hist aliases pre: hist is dead before pre is written)
    char* ws = (char*)d_ws;
    unsigned* bar  = (unsigned*)ws;                 // [0,    256)
    float* hbuf    = (float*)(ws + 256);            // [256,  4352)  2x512
    float* embed   = (float*)(ws + 4352);           // [4352, 6400)  512
    float* x       = (float*)(ws + 8192);           // 16384x512 f32 = 32MB
    float* pre     = (float*)(ws + 8192 + (size_t)N_NODES * 512 * 4); // 128MB
    float* hist    = pre;                           // alias, 32MB

    const size_t gemmLds = (size_t)32 * AS_LD * 4;  // 66KB dynamic LDS

    // 1) zero control block + h buffers + embed (1600 floats = 400 f4)
    zero_f4<<<2, 256, 0, stream>>>((float4*)ws, 400);
    // 2) zero histogram (8M floats = 2M float4)
    zero_f4<<<8192, 256, 0, stream>>>((float4*)hist, (long)N_NODES * FEAT / 4);
    // 3) node embedding gather -> x[:, 0:256]
    gather_emb<<<(N_NODES * (EMB / 4) + 255) / 256, 256, 0, stream>>>(
        node_tags, W_emb, b_emb, x);
    // 4) edge scatter-add into histogram
    edge_hist<<<(E + 255) / 256, 256, 0, stream>>>(nbr_nid, nbr_tag, hist, E);
    // 5) nb_emb = hist @ W_emb + b_emb -> x[:, 256:512]   (WMMA f32)
    {
        dim3 grid(N_NODES / 32, EMB / 128);
        wmma_gemm_f32<false><<<grid, 256, gemmLds, stream>>>(
            hist, W_emb, b_emb, x + EMB, 512, EMB);
    }
    // 6) pre = x @ W_ih^T + b_lstm   (WMMA f32, 34 GFLOP)
    {
        dim3 grid(N_NODES / 32, GATES / 128);
        wmma_gemm_f32<true><<<grid, 256, gemmLds, stream>>>(
            x, W_ih, b_lstm, pre, GATES, GATES);
    }
    // 7) sequential LSTM scan (persistent, W_hh sliced into LDS)
    {
        size_t smBytes = (size_t)(64 * 513 + 512 + 64 + 256 + 16 + 16) * 4;
        lstm_scan<<<NWG_LSTM, 256, smBytes, stream>>>(
            pre, W_hh, hbuf, embed, bar, N_NODES);
    }
    // 8) MLP head + log_softmax
    mlp_head<<<1, 256, 0, stream>>>(embed, W1, b1, W2, b2, (float*)d_out);
}